// DiverseBeamSearch_57234734187053
// MI455X (gfx1250) — compile-verified
//
#include <hip/hip_runtime.h>
#include <hip/hip_bf16.h>
#include <stdint.h>

// Diverse beam search step for MI455X (gfx1250).
//
// Roofline: lprobs is 32*12*50257 f32 = 77.2 MB read once -> ~3.3 us at
// 23.3 TB/s HBM. Everything else is KB-sized. So the kernel is a streaming
// top-k: we use the CDNA5 async copy engine (global_load_async_to_lds_b128,
// tracked by ASYNCcnt / s_wait_asynccnt) to double-buffer 4 KB tiles into LDS
// and scan them with ds_load_b128 readback. No matmul exists in this op, so
// WMMA is deliberately not used.
//
// Penalty trick: DIVERSITY_STRENGTH*div only LOWERS at most mini*(G-1)=9
// tokens per row, so the penalized top-3 of a row is contained in the raw
// per-row top-12. Phase 1 computes exact raw top-12 per (row, segment);
// Phase 2 (tiny) applies offsets + penalties to <=144 candidates per group
// and does the serial group logic, masking, and overlap update.

#define TOPK   12      // = mini*G = beam; superset bound mini + mini*(G-1)
#define PADTOK 1
#define NGROUP 4

// ---------- candidate ordering: value desc, then flattened index asc ----------
__device__ __forceinline__ bool cand_better(float va, int ia, float vb, int ib) {
  return (va > vb) || (va == vb && ia < ib);
}

__device__ __forceinline__ void insert_cand(float (&vals)[TOPK], int (&idxs)[TOPK],
                                            float v, int i) {
  if (!cand_better(v, i, vals[TOPK - 1], idxs[TOPK - 1])) return;
  vals[TOPK - 1] = v; idxs[TOPK - 1] = i;
#pragma unroll
  for (int k = TOPK - 1; k > 0; --k) {
    if (cand_better(vals[k], idxs[k], vals[k - 1], idxs[k - 1])) {
      float tv = vals[k]; vals[k] = vals[k - 1]; vals[k - 1] = tv;
      int   ti = idxs[k]; idxs[k] = idxs[k - 1]; idxs[k - 1] = ti;
    } else break;
  }
}

// merge two sorted-desc TOPK lists (b read-only) into a
__device__ __forceinline__ void merge_lists(float* av, int* ai,
                                            const float* bv, const int* bi) {
  float rv[TOPK]; int ri[TOPK];
  int i = 0, j = 0;
#pragma unroll
  for (int k = 0; k < TOPK; ++k) {
    bool ta;
    if (i >= TOPK)      ta = false;
    else if (j >= TOPK) ta = true;
    else                ta = cand_better(av[i], ai[i], bv[j], bi[j]);
    if (ta) { rv[k] = av[i]; ri[k] = ai[i]; ++i; }
    else    { rv[k] = bv[j]; ri[k] = bi[j]; ++j; }
  }
#pragma unroll
  for (int k = 0; k < TOPK; ++k) { av[k] = rv[k]; ai[k] = ri[k]; }
}

// ---------- CDNA5 async global->LDS copy engine (ASYNCcnt-tracked) ----------
// Per-lane: copies 16 bytes from global address gaddr to LDS offset lds_off, tracked by the wave's
// ASYNCcnt. Generic LDS pointers carry the wave-relative LDS byte offset in
// their low 32 bits (aperture in high dword), which is exactly what the
// VDST operand wants.
__device__ __forceinline__ void async_load_b128(unsigned lds_off, const float* gaddr) {
  unsigned long long ga = (unsigned long long)(uintptr_t)gaddr;
  asm volatile("global_load_async_to_lds_b128 %0, %1, off"
               :: "v"(lds_off), "v"(ga) : "memory");
}
__device__ __forceinline__ void wait_async_le1() {
  asm volatile("s_wait_asynccnt 1" ::: "memory");
}
__device__ __forceinline__ void wait_async_0() {
  asm volatile("s_wait_asynccnt 0" ::: "memory");
}

// =============================================================================
// Phase 1: raw top-12 of each (row, vocab-segment). grid = rows*split blocks.
// 256 threads/block (8 wave32). Streams the segment through double-buffered
// 4 KB LDS tiles via the async engine, per-thread sorted top-12, LDS tree
// merge, result to workspace.
// =============================================================================
__global__ __launch_bounds__(256) void topk_seg_kernel(
    const float* __restrict__ lprobs, int vocab, int seg, int split,
    float* __restrict__ wsV, int* __restrict__ wsI) {
  const int tid   = threadIdx.x;
  const int row   = blockIdx.x / split;
  const int s     = blockIdx.x - row * split;
  const int start = s * seg;
  int end = start + seg; if (end > vocab) end = vocab;
  const float* __restrict__ rp = lprobs + (size_t)row * (size_t)vocab;

  __shared__ __align__(16) float stage[2][1024];   // 2 x 4 KB tiles
  __shared__ float lv[256 * TOPK];
  __shared__ int   li[256 * TOPK];

  float vals[TOPK]; int idxs[TOPK];
#pragma unroll
  for (int k = 0; k < TOPK; ++k) { vals[k] = -__builtin_inff(); idxs[k] = 0x7fffffff; }

  const int len   = end - start;
  const int nFull = (len > 0) ? (len >> 10) : 0;   // full 1024-float tiles

  const unsigned ldsA = (unsigned)(uintptr_t)(&stage[0][tid << 2]);
  const unsigned ldsB = (unsigned)(uintptr_t)(&stage[1][tid << 2]);

  if (nFull > 0) {
    // prime pipeline
    async_load_b128(ldsA, rp + start + (tid << 2));
    for (int t = 0; t < nFull; ++t) {
      const int buf = t & 1;
      if (t + 1 < nFull) {
        async_load_b128(buf ? ldsA : ldsB,
                        rp + start + ((size_t)(t + 1) << 10) + (tid << 2));
        wait_async_le1();   // in-order completion => tile t landed
      } else {
        wait_async_0();
      }
      const float4 d = *reinterpret_cast<const float4*>(&stage[buf][tid << 2]);
      const int base = start + (t << 10) + (tid << 2);
      insert_cand(vals, idxs, d.x, base + 0);
      insert_cand(vals, idxs, d.y, base + 1);
      insert_cand(vals, idxs, d.z, base + 2);
      insert_cand(vals, idxs, d.w, base + 3);
    }
  }
  // tail (< 1024 floats) via plain global loads
  for (int i = start + (nFull << 10) + tid; i < end; i += 256)
    insert_cand(vals, idxs, rp[i], i);

  // block tree-merge of 256 sorted lists
#pragma unroll
  for (int k = 0; k < TOPK; ++k) { lv[tid * TOPK + k] = vals[k]; li[tid * TOPK + k] = idxs[k]; }
  __syncthreads();
  for (int off = 128; off > 0; off >>= 1) {
    if (tid < off)
      merge_lists(&lv[tid * TOPK], &li[tid * TOPK],
                  &lv[(tid + off) * TOPK], &li[(tid + off) * TOPK]);
    __syncthreads();
  }
  if (tid == 0) {
#pragma unroll
    for (int k = 0; k < TOPK; ++k) {
      wsV[(size_t)blockIdx.x * TOPK + k] = lv[k];
      wsI[(size_t)blockIdx.x * TOPK + k] = li[k];
    }
  }
}

// =============================================================================
// Phase 2: serial group logic per batch (one block/batch, lane 0 does the
// ~600-candidate bookkeeping). Applies score offsets, analytic diversity
// penalty, flattened-index tie-break, PAD masking, and the overlap update.
// =============================================================================
__global__ void finalize_kernel(
    const float* __restrict__ wsV, const int* __restrict__ wsI,
    const float* __restrict__ scores, const float* __restrict__ gov,
    const int* __restrict__ maskStop, const int* __restrict__ obi,
    const int* __restrict__ stepPtr, float* __restrict__ out,
    int bsz, int beam, int vocab, int T, int split) {
  if (threadIdx.x != 0) return;
  const int b    = blockIdx.x;
  const int G    = NGROUP;
  const int mini = beam / G;          // 3
  const int step = *stepPtr;
  const int ob   = obi[b];

  float ovin[NGROUP][NGROUP];
  for (int i = 0; i < G; ++i)
    for (int j = 0; j < G; ++j)
      ovin[i][j] = gov[((size_t)ob * G + i) * G + j];

  int lastTok[8][NGROUP];             // masked last tokens per (mini, group)
  const int perRow = split * TOPK;    // candidates per row (<= 48)
  const int nc     = mini * perRow;   // <= 144
  float cv[160]; int ctok[160]; int crow[160]; bool used[160];

  for (int g = 0; g < G; ++g) {
    // gather + penalize candidates
    for (int r = 0; r < mini; ++r) {
      const int beamIdx = g + r * G;
      const int row     = b * beam + beamIdx;
      const float off   = scores[(size_t)row * T + step];
      for (int q = 0; q < perRow; ++q) {
        const float v  = wsV[(size_t)row * perRow + q];
        const int  tok = wsI[(size_t)row * perRow + q];
        float div = 0.f;
        for (int gp = 0; gp < g; ++gp) {
          const float w = 1.f + ovin[g][gp];     // penalty weight (PAD included)
          for (int m = 0; m < mini; ++m)
            if (lastTok[m][gp] == tok) div += w;
        }
        const int c = r * perRow + q;
        cv[c]   = v + off - 0.5f * div;          // DIVERSITY_STRENGTH = -0.5
        ctok[c] = tok;
        crow[c] = r;
        used[c] = false;
      }
    }
    // select top-`mini` with top_k tie-break (value desc, flat index asc)
    for (int m = 0; m < mini; ++m) {
      int best = -1; float bv = 0.f; long long bflat = 0;
      for (int c = 0; c < nc; ++c) {
        if (used[c]) continue;
        const long long flat = (long long)crow[c] * vocab + ctok[c];
        if (best < 0 || cv[c] > bv || (cv[c] == bv && flat < bflat)) {
          best = c; bv = cv[c]; bflat = flat;
        }
      }
      used[best] = true;
      const int r = crow[best], tok = ctok[best];
      const int beamIdx = r * G + g;
      const int msk     = maskStop[b * beam + beamIdx];
      const int tokOut  = (msk == 0) ? PADTOK : tok;
      // outputs: scores_buf, indices_buf, beams_buf at [b, m*G+g]
      out[(size_t)b * beam + m * G + g]                          = bv;
      out[(size_t)bsz * beam + (size_t)b * beam + m * G + g]     = (float)tokOut;
      out[(size_t)2 * bsz * beam + (size_t)b * beam + m * G + g] = (float)beamIdx;
      lastTok[m][g] = tokOut;
    }
  }

  // new_group_overlap = overlap + 0.5 * group_overlap[obi]
  const size_t off3 = (size_t)3 * bsz * beam;
  for (int i = 0; i < G; ++i)
    for (int j = 0; j < G; ++j) {
      float cnt = 0.f;
      for (int m = 0; m < mini; ++m) {
        const int a = lastTok[m][i], c2 = lastTok[m][j];
        if (a != PADTOK && c2 != PADTOK && a == c2) cnt += 1.f;
      }
      out[off3 + ((size_t)b * G + i) * G + j] = cnt + 0.5f * ovin[i][j];
    }
}

extern "C" void kernel_launch(void* const* d_in, const int* in_sizes, int n_in,
                              void* d_out, int out_size, void* d_ws, size_t ws_size,
                              hipStream_t stream) {
  const float* lprobs = (const float*)d_in[0];
  const float* scores = (const float*)d_in[1];
  const float* gov    = (const float*)d_in[2];
  const int*   mask   = (const int*)d_in[3];
  // d_in[4] = prev_indices: provably unused (only its appended last token,
  // which is the freshly chosen masked token, feeds the outputs).
  const int*   obi    = (const int*)d_in[5];
  const int*   step   = (const int*)d_in[6];

  const int bsz   = in_sizes[5];            // 32
  const int bm    = in_sizes[3];            // bsz*beam = 384
  const int beam  = bm / bsz;               // 12
  const int vocab = in_sizes[0] / bm;       // 50257
  const int T     = in_sizes[1] / bm;       // 8
  const int rows  = bm;

  // split each row's vocab scan across segments for more blocks in flight
  int split = 4;
  size_t need = (size_t)rows * split * TOPK * (sizeof(float) + sizeof(int));
  if (ws_size < need) split = 1;
  const int seg = (vocab + split - 1) / split;

  float* wsV = (float*)d_ws;
  int*   wsI = (int*)((char*)d_ws + (size_t)rows * split * TOPK * sizeof(float));

  topk_seg_kernel<<<rows * split, 256, 0, stream>>>(lprobs, vocab, seg, split, wsV, wsI);
  finalize_kernel<<<bsz, 32, 0, stream>>>(wsV, wsI, scores, gov, mask, obi, step,
                                          (float*)d_out, bsz, beam, vocab, T, split);
}